// Word2Vec_47571057770740
// MI455X (gfx1250) — compile-verified
//
#include <hip/hip_runtime.h>
#include <hip/hip_bf16.h>

typedef __attribute__((ext_vector_type(2))) float v2f;
typedef __attribute__((ext_vector_type(8))) float v8f;

namespace {
constexpr int D = 128;              // d_model
constexpr int KNEG = 5;             // negatives per pair
constexpr int CHUNKS = D / 4;       // 32 WMMA K-steps (K=4 each)
constexpr int NCTX = KNEG + 1;      // pos + negatives = 6 score columns
constexpr int PAIRS_PER_WAVE = 16;  // one 16x16 WMMA tile of pairs per wave
constexpr int WAVES_PER_BLOCK = 8;  // 256 threads
constexpr int PAIRS_PER_BLOCK = PAIRS_PER_WAVE * WAVES_PER_BLOCK;  // 128
}

__device__ __forceinline__ float log_sigmoid_f(float z) {
    // numerically stable: log_sigmoid(z) = min(z,0) - log1p(exp(-|z|))
    return fminf(z, 0.0f) - log1pf(__expf(-fabsf(z)));
}

__global__ __launch_bounds__(256) void sgns_loss_kernel(
    const float* __restrict__ emb,      // [V, D]
    const float* __restrict__ out_w,    // [V, D]
    const int*   __restrict__ tgt_ids,  // [N]
    const int*   __restrict__ ctx_ids,  // [N]
    const int*   __restrict__ neg_ids,  // [N, KNEG]
    float*       __restrict__ loss,     // [N]
    int n)
{
    const int lane = threadIdx.x & 31;
    const int wave = threadIdx.x >> 5;
    const int base = (blockIdx.x * WAVES_PER_BLOCK + wave) * PAIRS_PER_WAVE;
    if (base >= n) return;              // wave-uniform; EXEC stays all-1s for WMMA

    const int m = lane & 15;            // matrix row (pair within tile) this lane feeds
    const int h = lane >> 4;            // half-wave selects K-pair within each chunk
    const int pair = base + m;

    // ---- preload ALL gather indices up front (no id-load stall inside the
    //      compute pipeline; their latency hides under the A-tile loads) ----
    const int tid = tgt_ids[pair];
    int ids[NCTX];
    ids[0] = ctx_ids[pair];
#pragma unroll
    for (int k = 0; k < KNEG; ++k)
        ids[k + 1] = neg_ids[pair * KNEG + k];   // 4B-aligned only; keep b32 loads

    // ---- A = emb rows of the 16 targets, in WMMA 16x4xf32 chunk layout ----
    // chunk c, lane L: holds X[m][4c+2h] , X[m][4c+2h+1]  (contiguous float2)
    const float* aptr = emb + (size_t)tid * D + 2 * h;
    v2f A[CHUNKS];
#pragma unroll
    for (int c = 0; c < CHUNKS; ++c)
        A[c] = *(const v2f*)(aptr + 4 * c);

    const int k7  = lane & 7;
    const int src = (m < 8) ? m : (m + 16);   // where diag element (m,m) lives

    float score[NCTX];

    // Fully unrolled: indices are compile-time constants (no scratch), and the
    // compiler can issue B-load clauses of context j+1 under the WMMAs of j.
#pragma unroll
    for (int j = 0; j < NCTX; ++j) {
        // B[k][n] = Y[n][k]: lane L holds Y[m][4c+2h], Y[m][4c+2h+1] — same layout as A
        const float* bptr = out_w + (size_t)ids[j] * D + 2 * h;

        v8f acc0 = {};   // even chunks
        v8f acc1 = {};   // odd chunks (break the WMMA RAW chain)
#pragma unroll
        for (int c = 0; c < CHUNKS; c += 2) {
            const v2f b0 = *(const v2f*)(bptr + 4 * c);
            const v2f b1 = *(const v2f*)(bptr + 4 * c + 4);
            acc0 = __builtin_amdgcn_wmma_f32_16x16x4_f32(
                false, A[c],     false, b0, (short)0, acc0, false, false);
            acc1 = __builtin_amdgcn_wmma_f32_16x16x4_f32(
                false, A[c + 1], false, b1, (short)0, acc1, false, false);
        }

        // ---- extract diagonal of S = X*Y^T ----
        // C layout: lanes 0-15 hold (M=v, N=lane); lanes 16-31 hold (M=8+v, N=lane-16).
        // diag(n,n): n<8 -> lane n, vgpr n ; n>=8 -> lane n+16, vgpr n-8.
        float dv = acc0[0] + acc1[0];
        dv = (k7 == 1) ? acc0[1] + acc1[1] : dv;
        dv = (k7 == 2) ? acc0[2] + acc1[2] : dv;
        dv = (k7 == 3) ? acc0[3] + acc1[3] : dv;
        dv = (k7 == 4) ? acc0[4] + acc1[4] : dv;
        dv = (k7 == 5) ? acc0[5] + acc1[5] : dv;
        dv = (k7 == 6) ? acc0[6] + acc1[6] : dv;
        dv = (k7 == 7) ? acc0[7] + acc1[7] : dv;

        score[j] = __shfl(dv, src, 32);   // lane m gets score of pair base+m
    }

    float total = log_sigmoid_f(score[0]);   // positive sample
#pragma unroll
    for (int j = 1; j < NCTX; ++j)
        total += log_sigmoid_f(-score[j]);   // negatives

    if (lane < 16)
        loss[pair] = -total;
}

extern "C" void kernel_launch(void* const* d_in, const int* in_sizes, int n_in,
                              void* d_out, int out_size, void* d_ws, size_t ws_size,
                              hipStream_t stream) {
    const float* emb   = (const float*)d_in[0];
    const float* out_w = (const float*)d_in[1];
    const int*   tgt   = (const int*)d_in[2];
    const int*   ctx   = (const int*)d_in[3];
    const int*   neg   = (const int*)d_in[4];
    float*       loss  = (float*)d_out;

    const int n = in_sizes[2];   // number of pairs
    const int blocks = (n + PAIRS_PER_BLOCK - 1) / PAIRS_PER_BLOCK;
    sgns_loss_kernel<<<blocks, 256, 0, stream>>>(emb, out_w, tgt, ctx, neg, loss, n);
}